// FrameAugment_39771397161403
// MI455X (gfx1250) — compile-verified
//
#include <hip/hip_runtime.h>
#include <cstdint>

// Problem constants (match reference: B=64, T=512, Fd=128)
#define TT   512
#define BB   64
#define FD   128
#define NB   64        // Cholesky panel width
#define SST  516       // padded LDS row stride (floats) for the S/P buffer

typedef __attribute__((ext_vector_type(2))) float v2f;
typedef __attribute__((ext_vector_type(8))) float v8f;

// ---------------------------------------------------------------------------
// wave32 reductions
// ---------------------------------------------------------------------------
__device__ __forceinline__ float wave_max(float v) {
#pragma unroll
    for (int o = 16; o > 0; o >>= 1) v = fmaxf(v, __shfl_xor(v, o, 32));
    return v;
}
__device__ __forceinline__ float wave_sum(float v) {
#pragma unroll
    for (int o = 16; o > 0; o >>= 1) v += __shfl_xor(v, o, 32);
    return v;
}

// ---------------------------------------------------------------------------
// Cholesky stage.  L lives in workspace (512x512 f32 = 1 MB).
// ---------------------------------------------------------------------------
__global__ void chol_init(const float* __restrict__ cov, float* __restrict__ L) {
    int idx = blockIdx.x * blockDim.x + threadIdx.x;
    if (idx >= TT * TT) return;
    int i = idx / TT, j = idx % TT;
    float v = 0.0f;                       // upper triangle must be exactly 0
    if (j <= i) {
        v = cov[idx];
        if (i == j) v += 1e-5f;
    }
    L[idx] = v;
}

// Factor one 64-wide panel (columns [p*NB, p*NB+NB)); prior SYRK updates
// already applied.  One workgroup, 512 threads, thread = matrix row.
__global__ void chol_panel(float* __restrict__ L, int p) {
    __shared__ float col[TT];
    __shared__ float dsh;
    const int tid  = threadIdx.x;          // row index 0..511
    const int kend = p * NB + NB;
    for (int k = p * NB; k < kend; ++k) {
        if (tid == k) {
            float d = sqrtf(L[k * TT + k]);
            L[k * TT + k] = d;
            dsh = d;
        }
        __syncthreads();
        const float dinv = 1.0f / dsh;
        float v = 0.0f;
        if (tid > k) {
            v = L[tid * TT + k] * dinv;
            L[tid * TT + k] = v;
        }
        col[tid] = v;
        __syncthreads();
        if (tid > k) {
            const int jmax = (tid < kend - 1) ? tid : (kend - 1);
            for (int j = k + 1; j <= jmax; ++j)
                L[tid * TT + j] -= v * col[j];
        }
        __syncthreads();
    }
}

// Trailing update: for all lower-triangle 64x64 tiles below the panel,
// A[I,J] -= panel_I * panel_J^T.  Fully parallel across blocks.
__global__ void chol_syrk(float* __restrict__ L, int p) {
    __shared__ float Ap[NB][NB + 1];
    __shared__ float Bp[NB][NB + 1];
    // triangular decode of blockIdx.x -> (it, jt), jt <= it
    int bid = blockIdx.x, it = 0;
    while (bid > it) { bid -= (it + 1); ++it; }
    const int jt  = bid;
    const int r0  = (p + 1) * NB + it * NB;
    const int c0  = (p + 1) * NB + jt * NB;
    const int kc  = p * NB;
    const int tid = threadIdx.x;           // 256 threads
    for (int e = tid; e < NB * NB; e += 256) {
        int i = e >> 6, k = e & 63;
        Ap[i][k] = L[(r0 + i) * TT + kc + k];
        Bp[i][k] = L[(c0 + i) * TT + kc + k];
    }
    __syncthreads();
    const int tx = tid & 63;               // output row within tile
    const int ty = tid >> 6;               // 16-col group 0..3
    float acc[16];
#pragma unroll
    for (int jj = 0; jj < 16; ++jj) acc[jj] = 0.0f;
    for (int k = 0; k < NB; ++k) {
        const float a = Ap[tx][k];
#pragma unroll
        for (int jj = 0; jj < 16; ++jj) acc[jj] += a * Bp[ty * 16 + jj][k];
    }
    const int gi = r0 + tx;
#pragma unroll
    for (int jj = 0; jj < 16; ++jj) {
        const int gj = c0 + ty * 16 + jj;
        if (gj <= gi) L[gi * TT + gj] -= acc[jj];   // keep upper triangle 0
    }
}

// ---------------------------------------------------------------------------
// Fused kernel: per (b, 64-row t-tile)
//   S = Z_tile @ L^T  (WMMA f32 16x16x4, triangular k-skip)
//   masked softmax over i (zeros for i>t handled analytically)
//   out_tile = P @ feature[b]  (WMMA f32)
// grid = (8, 64), block = 256 (8 waves), dynamic LDS = 64*SST*4 bytes
// ---------------------------------------------------------------------------
__global__ void fused_path_mix(const float* __restrict__ z,
                               const float* __restrict__ Lm,
                               const float* __restrict__ feat,
                               float* __restrict__ out) {
    extern __shared__ float S[];           // 64 rows x SST floats
    const int tb   = blockIdx.x;           // t-tile 0..7
    const int b    = blockIdx.y;           // batch 0..63
    const int t0   = tb * 64;
    const int tid  = threadIdx.x;
    // Wave index is wave-uniform: force it into an SGPR so all loop bounds
    // below are scalar (s_cbranch loops instead of EXEC-masked vector loops).
    const int wave = __builtin_amdgcn_readfirstlane(tid) >> 5;
    const int lane = tid & 31;
    const int mt   = wave & 3;             // 16-row M tile within the 64 rows
    const int half = wave >> 2;            // 0/1: splits the N range
    const int lr   = lane & 15;            // M (A) / N (B,C,D) lane index
    const int kq   = (lane >> 4) << 1;     // K pair selector: 0 or 2

    // ----- Pass 1: S tiles via WMMA -----
    const int ntiles = (tb + 1) * 4;       // compute columns i < 16*ntiles
    const int arow   = t0 + mt * 16 + lr;  // global t row for A fragments
    const float* zp  = z + ((size_t)arow * BB + b) * TT + kq;
    const int rbase  = mt * 16 + ((lane < 16) ? 0 : 8);

    for (int nt = half; nt < ntiles; nt += 2) {   // interleave halves (balance)
        const int n0 = nt * 16;
        const float* lp = Lm + (size_t)(n0 + lr) * TT + kq;
        v8f acc = {};
        const int kkmax = 4 * (nt + 1);    // k <= n0+15 (L strictly lower)
        // 4x software pipeline: batch 8 loads ahead of 4 WMMAs so the
        // s_wait_loadcnt becomes incremental instead of a full stall.
        for (int kk = 0; kk < kkmax; kk += 4) {
            v2f a[4], bf[4];
#pragma unroll
            for (int u = 0; u < 4; ++u) {
                a[u]  = *(const v2f*)(zp + (kk + u) * 4);
                bf[u] = *(const v2f*)(lp + (kk + u) * 4);
            }
#pragma unroll
            for (int u = 0; u < 4; ++u)
                acc = __builtin_amdgcn_wmma_f32_16x16x4_f32(
                    false, a[u], false, bf[u], (short)0, acc, false, false);
        }
        const int col = n0 + lr;
#pragma unroll
        for (int v = 0; v < 8; ++v)
            S[(rbase + v) * SST + col] = acc[v];
    }
    __syncthreads();

    // ----- masked softmax per row (8 rows per wave) -----
    for (int r = wave; r < 64; r += 8) {
        const int t = t0 + r;
        float* row = S + r * SST;
        float mx = 0.0f;                   // zero entries participate
        for (int i = lane; i <= t; i += 32) mx = fmaxf(mx, row[i]);
        mx = wave_max(mx);
        float s = 0.0f;
        for (int i = lane; i <= t; i += 32) {
            float e = __expf(row[i] - mx);
            row[i] = e;
            s += e;
        }
        s = wave_sum(s);
        const float qe = __expf(-mx);
        s += qe * (float)(511 - t);        // the (T-1-t) zero-padded logits
        const float inv = 1.0f / s;
        const float q = qe * inv;
        for (int i = lane; i < TT; i += 32)
            row[i] = (i <= t) ? row[i] * inv : q;
    }
    __syncthreads();

    // ----- Pass 2: out = P @ feature[b] via WMMA -----
    const float* fb   = feat + (size_t)b * TT * FD;
    const float* prow = S + (mt * 16 + lr) * SST + kq;   // A frags from LDS
    for (int nt = half * 4; nt < half * 4 + 4; ++nt) {
        const int n = nt * 16 + lr;
        v8f acc = {};
        for (int kk = 0; kk < TT / 4; kk += 4) {
            v2f a[4], bf[4];
#pragma unroll
            for (int u = 0; u < 4; ++u) {
                const int k = (kk + u) * 4 + kq;
                a[u] = *(const v2f*)(prow + (kk + u) * 4);
                bf[u].x = fb[(size_t)k * FD + n];
                bf[u].y = fb[(size_t)(k + 1) * FD + n];
            }
#pragma unroll
            for (int u = 0; u < 4; ++u)
                acc = __builtin_amdgcn_wmma_f32_16x16x4_f32(
                    false, a[u], false, bf[u], (short)0, acc, false, false);
        }
#pragma unroll
        for (int v = 0; v < 8; ++v) {
            const int t = t0 + rbase + v;
            out[((size_t)b * TT + t) * FD + n] = acc[v];
        }
    }
}

// ---------------------------------------------------------------------------
extern "C" void kernel_launch(void* const* d_in, const int* in_sizes, int n_in,
                              void* d_out, int out_size, void* d_ws, size_t ws_size,
                              hipStream_t stream) {
    const float* feature = (const float*)d_in[0];  // (B,T,F)
    const float* cov     = (const float*)d_in[1];  // (T,T)
    const float* z       = (const float*)d_in[2];  // (T,B,T)
    float* outp          = (float*)d_out;
    float* Lws           = (float*)d_ws;           // 512*512*4 = 1 MB scratch

    // Cholesky: init, then 8 blocked right-looking iterations
    chol_init<<<(TT * TT + 255) / 256, 256, 0, stream>>>(cov, Lws);
    for (int p = 0; p < TT / NB; ++p) {
        chol_panel<<<1, TT, 0, stream>>>(Lws, p);
        const int tr = (TT / NB) - 1 - p;          // trailing 64-blocks
        if (tr > 0)
            chol_syrk<<<tr * (tr + 1) / 2, 256, 0, stream>>>(Lws, p);
    }

    // Fused noise-GEMM + masked softmax + mix-GEMM
    dim3 grid(TT / 64, BB);
    const size_t smem = (size_t)64 * SST * sizeof(float);  // ~129 KB of 320 KB LDS
    fused_path_mix<<<grid, 256, smem, stream>>>(z, Lws, feature, outp);
}